// LossFunc_12421045420886
// MI455X (gfx1250) — compile-verified
//
#include <hip/hip_runtime.h>

#define BB 256
#define TT 200
#define QQ 1024
#define NROWS (BB * (TT - 1))      // 50944 rows = 6368 blocks * 8 waves
#define ROWS_PER_BLOCK 8

typedef __attribute__((ext_vector_type(2))) float v2f;
typedef __attribute__((ext_vector_type(4))) float f4;
typedef __attribute__((ext_vector_type(8))) float v8f;

// Exact wave32 sum-reduction via V_WMMA_F32_16X16X4_F32.
// A: VGPR0 = lane partial (K=0 lanes 0-15, K=2 lanes 16-31), VGPR1 = 0.
// B: all-ones (layout agnostic). C = 0.  => D[m,n] = x[lane m] + x[lane m+16].
// Per-lane sum of the 8 D VGPRs + one shfl_xor(16) yields the full wave sum
// in every lane.
__device__ __forceinline__ float wave_reduce_wmma(float x) {
    v2f a;     a[0] = x;    a[1] = 0.0f;
    v2f bones; bones[0] = 1.0f; bones[1] = 1.0f;
    v8f c = {};
    v8f d = __builtin_amdgcn_wmma_f32_16x16x4_f32(
        /*neg_a=*/false, a, /*neg_b=*/false, bones,
        /*c_mod=*/(short)0, c, /*reuse_a=*/false, /*reuse_b=*/false);
    float s = d[0] + d[1] + d[2] + d[3] + d[4] + d[5] + d[6] + d[7];
    s += __shfl_xor(s, 16, 32);
    return s;   // full wave sum in every lane
}

// One wave per (b,t) row: 8 independent rows per 256-thread block.
// Each lane issues 24 non-temporal b128 loads (8x pred, 8x batch-lo, 8x
// batch-hi) -> deep MLP, no barriers, no LDS staging.
__global__ void __launch_bounds__(256)
row_ll_kernel(const float* __restrict__ pred,
              const float* __restrict__ batch,
              float* __restrict__ ws) {
    const int wave = threadIdx.x >> 5;
    const int lane = threadIdx.x & 31;
    const int row  = blockIdx.x * ROWS_PER_BLOCK + wave;   // 0 .. NROWS-1
    const int b    = row / (TT - 1);
    const int t    = row - b * (TT - 1);                   // 0 .. T-2
    const size_t pred_off  = ((size_t)b * TT + t) * QQ;
    const size_t batch_off = ((size_t)b * TT + (t + 1)) * (size_t)(2 * QQ);

    float dot_p  = 0.0f;
    float diff_p = 0.0f;
#pragma unroll
    for (int k = 0; k < 8; ++k) {
        const int q = lane * 4 + k * 128;   // 32 lanes * 4 floats * 8 = 1024
        const f4 p  = __builtin_nontemporal_load((const f4*)(pred  + pred_off  + q));
        const f4 c1 = __builtin_nontemporal_load((const f4*)(batch + batch_off + q));
        const f4 c2 = __builtin_nontemporal_load((const f4*)(batch + batch_off + QQ + q));
        const f4 s  = c1 + c2;
        const f4 e  = c1 - c2;
        dot_p  += p.x * s.x + p.y * s.y + p.z * s.z + p.w * s.w;
        diff_p += e.x + e.y + e.z + e.w;
    }

    // WMMA wave reductions (EXEC all-1s: no divergence up to here).
    const float probs = wave_reduce_wmma(dot_p);
    const float sdiff = wave_reduce_wmma(diff_p);

    if (lane == 0) {
        const float a    = floorf((sdiff + 1.0f) * 0.5f);   // == correct bit
        const bool  mask = probs > 0.0f;
        const float pp   = mask ? probs : 0.5f;
        const float ll   = a * logf(pp) + (1.0f - a) * log1pf(-pp);
        ws[row] = mask ? ll : 0.0f;
    }
}

__global__ void __launch_bounds__(256)
final_reduce_kernel(const float* __restrict__ ws, float* __restrict__ out) {
    const int tid = threadIdx.x;
    float s = 0.0f;
    for (int i = tid; i < NROWS; i += 256) s += ws[i];   // NROWS % 256 == 0

    const float wsum = wave_reduce_wmma(s);

    __shared__ float sm[8];
    const int lane = tid & 31;
    const int wave = tid >> 5;
    if (lane == 0) sm[wave] = wsum;
    __syncthreads();

    if (tid == 0) {
        float tot = 0.0f;
#pragma unroll
        for (int w = 0; w < 8; ++w) tot += sm[w];
        out[0] = -tot;
    }
}

extern "C" void kernel_launch(void* const* d_in, const int* in_sizes, int n_in,
                              void* d_out, int out_size, void* d_ws, size_t ws_size,
                              hipStream_t stream) {
    const float* pred  = (const float*)d_in[0];   // (B, T, Q) fp32
    const float* batch = (const float*)d_in[1];   // (B, T, 2Q) fp32
    float* ws  = (float*)d_ws;                    // NROWS floats of scratch
    float* out = (float*)d_out;                   // 1 float

    row_ll_kernel<<<NROWS / ROWS_PER_BLOCK, 256, 0, stream>>>(pred, batch, ws);
    final_reduce_kernel<<<1, 256, 0, stream>>>(ws, out);
}